// ColRepeatCausalLinear_12481174962955
// MI455X (gfx1250) — compile-verified
//
#include <hip/hip_runtime.h>
#include <cstdint>

// out[b,e,t] = bias[t] + v[t] * y[b,e,t],  y[t] = d*y[t-1] + x[t],  d = clip(decay,0.9,1)
// Memory-bound scan (128 MiB traffic -> ~5.5us @ 23.3 TB/s). GEMM/WMMA formulation
// would be ~68.7 GFLOP of fp32 through the weakest WMMA shape (16x16x4) -> 100x slower.
// gfx1250-specific data path: global_load_async_to_lds_b128 + s_wait_asynccnt + ds_load.

#define AS1 __attribute__((address_space(1)))
#define AS3 __attribute__((address_space(3)))

typedef int v4i __attribute__((vector_size(16)));   // matches builtin's int4 pointee

#if __has_builtin(__builtin_amdgcn_global_load_async_to_lds_b128)
#define ASYNC_VIA_BUILTIN 1
#else
#define ASYNC_VIA_BUILTIN 0
#endif

__device__ __forceinline__ void async_load_b128(const float* gsrc, float* lds_dst) {
#if ASYNC_VIA_BUILTIN
  __builtin_amdgcn_global_load_async_to_lds_b128(
      (AS1 v4i*)(uintptr_t)gsrc,
      (AS3 v4i*)(uint32_t)(uintptr_t)lds_dst, 0, 0);
#else
  uint32_t loff = (uint32_t)(uintptr_t)lds_dst;      // low 32 bits of generic LDS ptr = LDS byte offset
  uint64_t gaddr = (uint64_t)(uintptr_t)gsrc;
  asm volatile("global_load_async_to_lds_b128 %0, %1, off"
               :: "v"(loff), "v"(gaddr) : "memory");
#endif
}

__device__ __forceinline__ void wait_async0() {
#if __has_builtin(__builtin_amdgcn_s_wait_asynccnt)
  __builtin_amdgcn_s_wait_asynccnt(0);
#else
  asm volatile("s_wait_asynccnt 0" ::: "memory");
#endif
}

__global__ __launch_bounds__(256)
void decayed_scan_kernel(const float* __restrict__ x,
                         const float* __restrict__ w,
                         const float* __restrict__ bias,
                         const float* __restrict__ decay,
                         float* __restrict__ out,
                         int S) {
  __shared__ float xs[2048];   // one row staged via async loads
  __shared__ float wavetot[8]; // per-wave inclusive totals

  const int tid  = threadIdx.x;
  const int lane = tid & 31;
  const int wid  = tid >> 5;
  const long long row = blockIdx.x;

  const float d = fminf(fmaxf(decay[0], 0.9f), 1.0f);

  const float* xrow = x + row * (long long)S + tid * 8;

  // ---- stage this thread's 8 elements into LDS via async global->LDS ----
  async_load_b128(xrow,     &xs[tid * 8]);
  async_load_b128(xrow + 4, &xs[tid * 8 + 4]);
  wait_async0();
  __asm__ volatile("" ::: "memory");

  float4 a  = *(const float4*)&xs[tid * 8];
  float4 b4 = *(const float4*)&xs[tid * 8 + 4];

  // ---- serial decayed scan over the 8 local elements ----
  float l0 = a.x;
  float l1 = fmaf(d, l0, a.y);
  float l2 = fmaf(d, l1, a.z);
  float l3 = fmaf(d, l2, a.w);
  float l4 = fmaf(d, l3, b4.x);
  float l5 = fmaf(d, l4, b4.y);
  float l6 = fmaf(d, l5, b4.z);
  float l7 = fmaf(d, l6, b4.w);

  const float d2 = d * d;
  const float d3 = d2 * d;
  const float d4 = d2 * d2;
  const float d5 = d4 * d;
  const float d6 = d4 * d2;
  const float d7 = d6 * d;
  const float r8 = d4 * d4;   // decay across one 8-element segment

  // ---- wave32 Kogge-Stone decayed scan over per-thread segment totals ----
  float v = l7;
  float scale = r8;           // r8^(1), then r8^2, r8^4, r8^8, r8^16
  #pragma unroll
  for (int off = 1; off < 32; off <<= 1) {
    float up = __shfl_up(v, (unsigned)off, 32);
    if (lane >= off) v = fmaf(scale, up, v);
    scale = scale * scale;
  }
  // v = decayed inclusive sum ending at this thread's last element (within wave)

  if (lane == 31) wavetot[wid] = v;
  __syncthreads();

  // ---- cross-wave carry (8 waves, ratio d^256) ----
  const float r16 = r8 * r8, r32 = r16 * r16, r64 = r32 * r32,
              r128 = r64 * r64, r256 = r128 * r128;
  float carry = 0.f;
  for (int u = 0; u < wid; ++u) carry = fmaf(carry, r256, wavetot[u]);

  // exclusive-within-wave value, then add decayed wave carry
  float excl = __shfl_up(v, 1u, 32);
  if (lane == 0) excl = 0.f;
  const float pre = fmaf(carry, __powf(r8, (float)lane), excl); // y at (segment start - 1)

  // ---- finalize y per element: y_i = l_i + pre * d^(i+1) ----
  const float y0 = fmaf(pre, d,  l0);
  const float y1 = fmaf(pre, d2, l1);
  const float y2 = fmaf(pre, d3, l2);
  const float y3 = fmaf(pre, d4, l3);
  const float y4 = fmaf(pre, d5, l4);
  const float y5 = fmaf(pre, d6, l5);
  const float y6 = fmaf(pre, d7, l6);
  const float y7 = fmaf(pre, r8, l7);

  // ---- epilogue: out = bias + v * y ----
  const float4 w0 = *(const float4*)(w + tid * 8);
  const float4 w1 = *(const float4*)(w + tid * 8 + 4);
  const float4 bb0 = *(const float4*)(bias + tid * 8);
  const float4 bb1 = *(const float4*)(bias + tid * 8 + 4);

  float4 o0, o1;
  o0.x = fmaf(w0.x, y0, bb0.x);
  o0.y = fmaf(w0.y, y1, bb0.y);
  o0.z = fmaf(w0.z, y2, bb0.z);
  o0.w = fmaf(w0.w, y3, bb0.w);
  o1.x = fmaf(w1.x, y4, bb1.x);
  o1.y = fmaf(w1.y, y5, bb1.y);
  o1.z = fmaf(w1.z, y6, bb1.z);
  o1.w = fmaf(w1.w, y7, bb1.w);

  float* orow = out + row * (long long)S + tid * 8;
  *(float4*)(orow)     = o0;
  *(float4*)(orow + 4) = o1;
}

extern "C" void kernel_launch(void* const* d_in, const int* in_sizes, int n_in,
                              void* d_out, int out_size, void* d_ws, size_t ws_size,
                              hipStream_t stream) {
  const float* x     = (const float*)d_in[0];
  const float* w     = (const float*)d_in[1];   // (1,S) -> S floats
  const float* bias  = (const float*)d_in[2];   // S floats
  const float* decay = (const float*)d_in[3];   // 1 float
  float* out = (float*)d_out;

  const int S = in_sizes[1];                        // 2048
  const long long rows = (long long)in_sizes[0] / S; // B*E = 8192

  dim3 grid((unsigned)rows);
  dim3 block(256);   // S/8 threads, 8 waves (wave32)
  hipLaunchKernelGGL(decayed_scan_kernel, grid, block, 0, stream,
                     x, w, bias, decay, out, S);
}